// BoundaryLoss_81630148428070
// MI455X (gfx1250) — compile-verified
//
#include <hip/hip_runtime.h>
#include <hip/hip_bf16.h>

#define B_   4
#define C_   8
#define H_   256
#define W_   256
#define NMAP (B_ * C_)        // 32 (b,c) maps
#define HW   (H_ * W_)        // 65536
#define INF_ 512.0f           // H + W, matches reference

typedef __attribute__((ext_vector_type(2))) float v2f;
typedef __attribute__((ext_vector_type(4))) float v4f;
typedef __attribute__((ext_vector_type(8))) float v8f;
typedef __attribute__((ext_vector_type(4))) int   v4i;
typedef __attribute__((ext_vector_type(8))) int   v8i;

// ---------------------------------------------------------------- init
__global__ void k_init(float* __restrict__ bl, int* __restrict__ present) {
    int t = threadIdx.x;
    if (t < NMAP) { bl[t] = 0.0f; present[t] = 0; }
}

// ---------------------------------------------------------------- softmax over C (=8)
__global__ void k_softmax(const float* __restrict__ logits, float* __restrict__ probs) {
    int p = blockIdx.x * blockDim.x + threadIdx.x;   // pixel id in [0, B*H*W)
    if (p >= B_ * HW) return;
    int b = p / HW, pix = p - b * HW;
    const float* src = logits + (size_t)b * C_ * HW + pix;
    float v[C_];
    float mx = -3.4e38f;
    #pragma unroll
    for (int c = 0; c < C_; ++c) { v[c] = src[(size_t)c * HW]; mx = fmaxf(mx, v[c]); }
    float s = 0.0f;
    #pragma unroll
    for (int c = 0; c < C_; ++c) { v[c] = expf(v[c] - mx); s += v[c]; }
    float inv = 1.0f / s;
    float* dst = probs + (size_t)b * C_ * HW + pix;
    #pragma unroll
    for (int c = 0; c < C_; ++c) dst[(size_t)c * HW] = v[c] * inv;
}

// ---------------------------------------------------------------- 1-D row scans -> g^2
// blockIdx.x = map-polarity plane (0..63), threadIdx.x = row
__global__ void k_rowscan(const int* __restrict__ target, float* __restrict__ g2,
                          int* __restrict__ present) {
    int mp  = blockIdx.x;
    int m   = mp >> 1;
    int pol = mp & 1;                 // 0: sites = (target==c), 1: sites = (target!=c)
    int b   = m / C_;
    int c   = m - b * C_;
    int r   = threadIdx.x;            // row
    const int* trow = target + ((size_t)b * H_ + r) * W_;
    float*     grow = g2 + ((size_t)mp * H_ + r) * W_;

    // left pass: distance to nearest site on the left (incl. self); -INF sentinel
    float left = -INF_;
    int   found = 0;
    for (int j = 0; j < W_; ++j) {
        int site = ((trow[j] == c) == (pol == 0));
        if (site) { left = (float)j; found = 1; }
        grow[j] = (float)j - left;
    }
    if (pol == 0 && found) atomicOr(&present[m], 1);

    // right pass: combine with nearest site on the right, clamp to INF, square
    float right = 3.0f * INF_;
    for (int j = W_ - 1; j >= 0; --j) {
        int site = ((trow[j] == c) == (pol == 0));
        if (site) right = (float)j;
        float g = fminf(fminf(grow[j], right - (float)j), INF_);
        grow[j] = g * g;
    }
}

// ---------------------------------------------------------------- TDM descriptor builder
// D# group0/group1 per CDNA5 ISA.
// 2D tile: tile_dim0 = 16 elems (64B rows), tile_dim1 = 256 rows,
// tensor_dim0_stride = 256 elems, data_size = 4B. LDS packing is row-major
// contiguous (x fastest), matching float[256][16].
__device__ __forceinline__ v4i tdm_group0(unsigned lds_off, unsigned long long gaddr) {
    v4i g0;
    g0[0] = 1;                                    // count=1, is_restore=0, gather=0
    g0[1] = (int)lds_off;                         // lds_addr (bytes)
    g0[2] = (int)(unsigned)(gaddr & 0xffffffffull);          // global_addr[31:0]
    g0[3] = (int)(((unsigned)(gaddr >> 32) & 0x1ffffffu)     // global_addr[56:32]
                  | (2u << 30));                             // type=2 ("image")
    return g0;
}
__device__ __forceinline__ v8i tdm_group1() {
    v8i g1;
    g1[0] = 0x00020000;          // data_size=2 (4B); mask/flags/pad = 0
    g1[1] = (int)(256u << 16);   // tensor_dim0[15:0]=256 in bits[31:16]
    g1[2] = (int)(256u << 16);   // tensor_dim0[31:16]=0 | tensor_dim1[15:0]=256
    g1[3] = (int)(16u  << 16);   // tensor_dim1[31:16]=0 | tile_dim0=16
    g1[4] = 256;                 // tile_dim1=256, tile_dim2=0 (unused)
    g1[5] = 256;                 // tensor_dim0_stride[31:0] = 256
    g1[6] = 0;                   // stride hi / tensor_dim1_stride lo = 0
    g1[7] = 0;                   // tensor_dim1_stride hi = 0
    return g1;
}

// ---------------------------------------------------------------- column min-plus + fuse product
// blockIdx.x = m*16 + jtile; threads = rows (256). probs overwritten with probs*dmap.
// g^2 column tiles are staged into LDS by the Tensor Data Mover.
__global__ void k_colpass(const float* __restrict__ g2, float* __restrict__ probs) {
    int m  = blockIdx.x >> 4;
    int j0 = (blockIdx.x & 15) << 4;                 // 16-column tile
    const float* gp = g2 + ((size_t)(2 * m)     * HW);
    const float* gn = g2 + ((size_t)(2 * m + 1) * HW);

    __shared__ float stile[2][H_][16];               // [pol][k][jt], 32 KB

    if (threadIdx.x < 32) {                          // one wave issues the DMAs
        unsigned lds_p = (unsigned)(unsigned long long)(uintptr_t)(void*)&stile[0][0][0];
        unsigned lds_n = (unsigned)(unsigned long long)(uintptr_t)(void*)&stile[1][0][0];
        v4i g0p = tdm_group0(lds_p, (unsigned long long)(uintptr_t)(gp + j0));
        v4i g0n = tdm_group0(lds_n, (unsigned long long)(uintptr_t)(gn + j0));
        v8i g1  = tdm_group1();
        asm volatile("tensor_load_to_lds %0, %1" :: "s"(g0p), "s"(g1) : "memory");
        asm volatile("tensor_load_to_lds %0, %1" :: "s"(g0n), "s"(g1) : "memory");
        __builtin_amdgcn_s_wait_tensorcnt(0);
    }
    __syncthreads();

    int i = threadIdx.x;                             // this thread's row
    float accp[16], accn[16];
    #pragma unroll
    for (int jt = 0; jt < 16; ++jt) { accp[jt] = 3.4e38f; accn[jt] = 3.4e38f; }

    for (int k = 0; k < H_; ++k) {
        float t   = (float)(i - k);
        float tsq = t * t;
        #pragma unroll
        for (int jt = 0; jt < 16; ++jt) {
            accp[jt] = fminf(accp[jt], tsq + stile[0][k][jt]);
            accn[jt] = fminf(accn[jt], tsq + stile[1][k][jt]);
        }
    }

    float* prow = probs + ((size_t)m * H_ + i) * W_ + j0;
    #pragma unroll
    for (int jt = 0; jt < 16; ++jt) {
        float d = sqrtf(accp[jt]) - sqrtf(accn[jt]);  // edt(mask) - edt(~mask)
        prow[jt] = prow[jt] * d;                       // fused probs * dmap
    }
}

// ---------------------------------------------------------------- WMMA-ones reduction
// One block per map, 8 waves. Sum 65536 products via v_wmma_f32_16x16x4_f32 with B=1.
// b128 loads: each lane loads float4 -> 128 floats per wave-iter -> 2 WMMAs.
__global__ void k_reduce_wmma(const float* __restrict__ prod, float* __restrict__ bl) {
    int m    = blockIdx.x;
    int tid  = threadIdx.x;
    int lane = tid & 31;
    int wave = tid >> 5;
    const v4f* base = (const v4f*)(prod + (size_t)m * HW + (size_t)wave * (HW / 8));

    v8f acc = {0.f, 0.f, 0.f, 0.f, 0.f, 0.f, 0.f, 0.f};
    v2f ones; ones[0] = 1.0f; ones[1] = 1.0f;

    // 8192 elements per wave, 128 per iteration -> 64 iterations, 2 WMMAs each
    for (int it = 0; it < HW / (8 * 128); ++it) {
        v4f x = base[it * 32 + lane];
        v2f a0; a0[0] = x[0]; a0[1] = x[1];   // any bijection of 64 values into A
        v2f a1; a1[0] = x[2]; a1[1] = x[3];   // works: C[m,n] = sum_K A[m,K] * 1
        acc = __builtin_amdgcn_wmma_f32_16x16x4_f32(
            false, a0, false, ones, (short)0, acc, false, false);
        acc = __builtin_amdgcn_wmma_f32_16x16x4_f32(
            false, a1, false, ones, (short)0, acc, false, false);
    }

    // each C entry row-sum is replicated across 16 columns -> total = 16 * true sum
    float s = acc[0] + acc[1] + acc[2] + acc[3] + acc[4] + acc[5] + acc[6] + acc[7];
    for (int o = 16; o > 0; o >>= 1) s += __shfl_xor(s, o, 32);

    __shared__ float ws[8];
    if (lane == 0) ws[wave] = s;
    __syncthreads();
    if (tid == 0) {
        float t = 0.0f;
        for (int w = 0; w < 8; ++w) t += ws[w];
        bl[m] = t * (1.0f / 16.0f);
    }
}

// ---------------------------------------------------------------- finalize
__global__ void k_finalize(const float* __restrict__ bl, const int* __restrict__ present,
                           float* __restrict__ out) {
    if (threadIdx.x == 0 && blockIdx.x == 0) {
        float s = 0.0f, np = 0.0f;
        for (int m = 0; m < NMAP; ++m)
            if (present[m]) { s += bl[m]; np += 1.0f; }
        out[0] = s / fmaxf(np, 1.0f);
    }
}

// ---------------------------------------------------------------- launch
extern "C" void kernel_launch(void* const* d_in, const int* in_sizes, int n_in,
                              void* d_out, int out_size, void* d_ws, size_t ws_size,
                              hipStream_t stream) {
    const float* logits = (const float*)d_in[0];   // [B, C, H, W] f32
    const int*   target = (const int*)d_in[1];     // [B, H, W]   i32

    float* probs   = (float*)d_ws;                         // NMAP*HW floats (8 MB)
    float* g2      = probs + (size_t)NMAP * HW;            // 2*NMAP*HW floats (16 MB)
    float* bl      = g2 + (size_t)2 * NMAP * HW;           // 32 floats
    int*   present = (int*)(bl + NMAP);                    // 32 ints
    float* out     = (float*)d_out;

    k_init<<<1, 64, 0, stream>>>(bl, present);
    k_softmax<<<(B_ * HW + 255) / 256, 256, 0, stream>>>(logits, probs);
    k_rowscan<<<2 * NMAP, H_, 0, stream>>>(target, g2, present);
    k_colpass<<<NMAP * 16, 256, 0, stream>>>(g2, probs);
    k_reduce_wmma<<<NMAP, 256, 0, stream>>>(probs, bl);
    k_finalize<<<1, 1, 0, stream>>>(bl, present, out);
}